// Block_1975684956321
// MI455X (gfx1250) — compile-verified
//
#include <hip/hip_runtime.h>
#include <hip/hip_bf16.h>

typedef float v2f __attribute__((ext_vector_type(2)));
typedef float v8f __attribute__((ext_vector_type(8)));
typedef int   v4i __attribute__((ext_vector_type(4)));

#define D_ 64
#define H_ 4
#define ED_ 32

__device__ __forceinline__ float sigmoidf_(float x) { return 1.0f / (1.0f + __expf(-x)); }

__device__ __forceinline__ void atomicMaxF(float* addr, float v) {
    if (v >= 0.0f) atomicMax((int*)addr, __float_as_int(v));
    else           atomicMin((unsigned int*)addr, (unsigned int)__float_as_int(v));
}

// ---------------------------------------------------------------------------
// CDNA5 async global->LDS copy (16B per lane), with portable fallback.
// Toolchain declares the builtin with generic v4i* pointer params.
// ---------------------------------------------------------------------------
#if defined(__AMDGCN__) && __has_builtin(__builtin_amdgcn_global_load_async_to_lds_b128)
#define HAVE_ASYNC_LDS 1
#else
#define HAVE_ASYNC_LDS 0
#endif

__device__ __forceinline__ void async_copy_b128(const void* g, void* l) {
#if HAVE_ASYNC_LDS
    __builtin_amdgcn_global_load_async_to_lds_b128((v4i*)g, (v4i*)l, 0, 0);
#else
    *(float4*)l = *(const float4*)g;
#endif
}

__device__ __forceinline__ void async_wait0() {
#if defined(__AMDGCN__) && __has_builtin(__builtin_amdgcn_s_wait_asynccnt)
    __builtin_amdgcn_s_wait_asynccnt(0);
#elif HAVE_ASYNC_LDS
    asm volatile("s_wait_asynccnt 0x0" ::: "memory");
#endif
}

// ---------------------------------------------------------------------------
// Unified f32 WMMA GEMM: C[M,Ncols] = act(A[M,K] @ B + bias)
// TRANSB=false: B is [K,Ncols] row-major.  TRANSB=true: B is [Ncols,K] (use B^T).
// ACT: 0 = none, 1 = celu(alpha=1).
// Block = 128 threads (4 waves); wave w computes the 16x16 tile at column
// (blockIdx.y*4+w)*16. Both A (16x64) and the block's B slice (64x64) are
// staged into LDS with async b128 copies. All fragments of a K-chunk are
// hoisted into registers before the WMMA chain so the 16 WMMAs issue
// back-to-back behind a single dscnt wait.
// Requires M%16==0, K%64==0, Ncols%64==0 (true for all uses here).
// ---------------------------------------------------------------------------
template <bool TRANSB, int ACT, bool BIAS>
__global__ __launch_bounds__(128)
void gemm_wmma_f32(const float* __restrict__ A, const float* __restrict__ B,
                   const float* __restrict__ bias, float* __restrict__ C,
                   int M, int K, int Ncols)
{
    constexpr int BS = TRANSB ? 68 : 72;   // bank-conflict-free LDS strides
    __shared__ float Atile[16 * 68];
    __shared__ float Btile[64 * BS];

    const int tid  = threadIdx.x;
    const int wave = tid >> 5;
    const int lane = tid & 31;
    const int half = lane >> 4;      // A frag: K +2 | C frag: row +8
    const int ln16 = lane & 15;
    const int bm   = blockIdx.x;
    const int colbase = blockIdx.y * 64;
    const int col0 = colbase + wave * 16;
    const int ln   = wave * 16 + ln16;     // local column 0..63

    v8f acc = {};
    for (int kk = 0; kk < K; kk += 64) {
        // --- stage A chunk 16x64 (256 x 16B) ---
        #pragma unroll
        for (int i = 0; i < 2; ++i) {
            int idx = tid + i * 128;               // 0..255
            int r = idx >> 4, c4 = (idx & 15) * 4;
            async_copy_b128(A + (long)(bm * 16 + r) * K + kk + c4,
                            &Atile[r * 68 + c4]);
        }
        // --- stage B chunk 64x64 (1024 x 16B) ---
        #pragma unroll
        for (int i = 0; i < 8; ++i) {
            int idx = tid + i * 128;               // 0..1023
            if (TRANSB) {   // Btile[n][k]
                int n = idx >> 4, k4 = (idx & 15) * 4;
                async_copy_b128(B + (long)(colbase + n) * K + kk + k4,
                                &Btile[n * BS + k4]);
            } else {        // Btile[k][n]
                int k = idx >> 4, n4 = (idx & 15) * 4;
                async_copy_b128(B + (long)(kk + k) * Ncols + colbase + n4,
                                &Btile[k * BS + n4]);
            }
        }
        async_wait0();
        __syncthreads();

        // Hoist all fragments to registers, then run the WMMA chain.
        v2f afr[16], bfr[16];
        #pragma unroll
        for (int k4 = 0; k4 < 16; ++k4) {
            const int kb = k4 * 4 + 2 * half;
            afr[k4].x = Atile[ln16 * 68 + kb];
            afr[k4].y = Atile[ln16 * 68 + kb + 1];
            if (TRANSB) {
                bfr[k4].x = Btile[ln * BS + kb];
                bfr[k4].y = Btile[ln * BS + kb + 1];
            } else {
                bfr[k4].x = Btile[kb * BS + ln];
                bfr[k4].y = Btile[(kb + 1) * BS + ln];
            }
        }
        #pragma unroll
        for (int k4 = 0; k4 < 16; ++k4) {
            acc = __builtin_amdgcn_wmma_f32_16x16x4_f32(
                false, afr[k4], false, bfr[k4], (short)0, acc, false, false);
        }
        __syncthreads();
    }
    const float bv = BIAS ? bias[col0 + ln16] : 0.0f;
    #pragma unroll
    for (int r = 0; r < 8; ++r) {
        int row = bm * 16 + r + 8 * half;
        float v = acc[r] + bv;
        if (ACT == 1) v = v > 0.0f ? v : (__expf(v) - 1.0f);   // celu(1)
        C[(long)row * Ncols + col0 + ln16] = v;
    }
}

// ---------------------------------------------------------------------------
// init: x_cur = x, h = x
// ---------------------------------------------------------------------------
__global__ void k_init(const float* __restrict__ x, float* __restrict__ xcur,
                       float* __restrict__ h, long n)
{
    long i = (long)blockIdx.x * blockDim.x + threadIdx.x;
    if (i < n) { float v = x[i]; xcur[i] = v; h[i] = v; }
}

// w2e[h*32+k] = sum_d W_edge[k,h*64+d] * W_att[h, 64+d]   (one-time, tiny)
__global__ void k_w2e(const float* __restrict__ W_edge, const float* __restrict__ W_att,
                      float* __restrict__ w2e)
{
    int i = threadIdx.x;                 // 128 threads
    int h = i >> 5, k = i & 31;
    float acc = 0.0f;
    for (int d = 0; d < 64; ++d)
        acc += W_edge[k * 256 + h * 64 + d] * W_att[h * 192 + 64 + d];
    w2e[i] = acc;
}

// per-iteration reset: agg=0, amax=-inf, denom=0
__global__ void k_reset(float* __restrict__ agg, float* __restrict__ amax,
                        float* __restrict__ denom, int Nn)
{
    long i = (long)blockIdx.x * blockDim.x + threadIdx.x;
    if (i < (long)Nn * 256) agg[i] = 0.0f;
    if (i < (long)Nn * 4) { amax[i] = -__builtin_inff(); denom[i] = 0.0f; }
}

// ---------------------------------------------------------------------------
// Per-node score pre-reduction: s[i,h]=<xp[i,h,:],Wa1[h]>, t[i,h]=<xp[i,h,:],Wa3[h]>
// One wave per node; lane holds 8 elems (c = lane+32j, head = j>>1).
// ---------------------------------------------------------------------------
__global__ __launch_bounds__(256)
void k_score_pre(const float* __restrict__ xp, const float* __restrict__ W_att,
                 float* __restrict__ s, float* __restrict__ t, int Nn)
{
    int node = blockIdx.x * 8 + (threadIdx.x >> 5);
    if (node >= Nn) return;
    int lane = threadIdx.x & 31;
    const float* xr = xp + (long)node * 256;
    float sp[4] = {0, 0, 0, 0}, tp[4] = {0, 0, 0, 0};
    #pragma unroll
    for (int j = 0; j < 8; ++j) {
        int c = lane + 32 * j;
        int h = j >> 1;
        int d = c & 63;
        float xv = xr[c];
        sp[h] += xv * W_att[h * 192 + d];
        tp[h] += xv * W_att[h * 192 + 128 + d];
    }
    #pragma unroll
    for (int off = 16; off > 0; off >>= 1) {
        #pragma unroll
        for (int h = 0; h < 4; ++h) {
            sp[h] += __shfl_xor(sp[h], off, 32);
            tp[h] += __shfl_xor(tp[h], off, 32);
        }
    }
    if (lane == 0) {
        #pragma unroll
        for (int h = 0; h < 4; ++h) {
            s[(long)node * 4 + h] = sp[h];
            t[(long)node * 4 + h] = tp[h];
        }
    }
}

// ---------------------------------------------------------------------------
// Attention scores (thread per edge): alpha = leaky(s[dst]+t[src]+ea.w2e), atomic max.
// ---------------------------------------------------------------------------
__global__ __launch_bounds__(256)
void k_att(const int* __restrict__ ei, const float* __restrict__ ea,
           const float* __restrict__ s, const float* __restrict__ t,
           const float* __restrict__ w2e, float* __restrict__ alpha,
           float* __restrict__ amax, int En)
{
    int e = blockIdx.x * blockDim.x + threadIdx.x;
    if (e >= En) return;
    int src = ei[e], dst = ei[En + e];
    const float4* eav = (const float4*)(ea + (long)e * 32);
    float acc[4] = {0, 0, 0, 0};
    #pragma unroll
    for (int q = 0; q < 8; ++q) {
        float4 v = eav[q];
        #pragma unroll
        for (int h = 0; h < 4; ++h) {
            acc[h] += v.x * w2e[h * 32 + q * 4 + 0];
            acc[h] += v.y * w2e[h * 32 + q * 4 + 1];
            acc[h] += v.z * w2e[h * 32 + q * 4 + 2];
            acc[h] += v.w * w2e[h * 32 + q * 4 + 3];
        }
    }
    #pragma unroll
    for (int h = 0; h < 4; ++h) {
        float a = acc[h] + s[(long)dst * 4 + h] + t[(long)src * 4 + h];
        a = a > 0.0f ? a : 0.2f * a;          // leaky relu
        alpha[(long)e * 4 + h] = a;
        atomicMaxF(&amax[(long)dst * 4 + h], a);
    }
}

// exp pass: ex = exp(alpha - amax[dst]); denom += ex
__global__ __launch_bounds__(256)
void k_exp(const int* __restrict__ ei, float* __restrict__ alpha,
           const float* __restrict__ amax, float* __restrict__ denom, int En)
{
    int e = blockIdx.x * blockDim.x + threadIdx.x;
    if (e >= En) return;
    int dst = ei[En + e];
    #pragma unroll
    for (int h = 0; h < 4; ++h) {
        float ex = __expf(alpha[(long)e * 4 + h] - amax[(long)dst * 4 + h]);
        alpha[(long)e * 4 + h] = ex;
        atomicAdd(&denom[(long)dst * 4 + h], ex);
    }
}

// ---------------------------------------------------------------------------
// Message pass: wave per edge; W_edge (32x256 = 32KB) staged into LDS with
// async b128 copies; ep recomputed on the fly; msg = (ex/denom)*ep*xj
// scattered via f32 atomics (agg is L2-resident). Block = 64 edges.
// ---------------------------------------------------------------------------
#define MSG_EPB 64
__global__ __launch_bounds__(256)
void k_msg(const int* __restrict__ ei, const float* __restrict__ ea,
           const float* __restrict__ xp, const float* __restrict__ W_edge,
           const float* __restrict__ alpha, const float* __restrict__ denom,
           float* __restrict__ agg, int En)
{
    __shared__ float Wl[32 * 256];
    #pragma unroll
    for (int i = 0; i < 8; ++i) {
        int idx = threadIdx.x + i * 256;          // 0..2047 (16B chunks)
        async_copy_b128(W_edge + idx * 4, &Wl[idx * 4]);
    }
    async_wait0();
    __syncthreads();

    const int wave = threadIdx.x >> 5;
    const int lane = threadIdx.x & 31;
    const int base = blockIdx.x * MSG_EPB;

    for (int it = 0; it < MSG_EPB / 8; ++it) {
        int e = base + it * 8 + wave;
        if (e >= En) break;
        // prefetch next edge's attributes for this wave (global_prefetch_b8)
        if (e + 8 < En) __builtin_prefetch(ea + (long)(e + 8) * 32 + lane, 0, 1);

        int src = ei[e], dst = ei[En + e];
        float eareg = ea[(long)e * 32 + lane];
        float w4[4];
        #pragma unroll
        for (int h = 0; h < 4; ++h)
            w4[h] = alpha[(long)e * 4 + h] / (denom[(long)dst * 4 + h] + 1e-16f);

        float ep[8] = {0, 0, 0, 0, 0, 0, 0, 0};
        #pragma unroll 8
        for (int k = 0; k < 32; ++k) {
            float eak = __shfl(eareg, k, 32);
            #pragma unroll
            for (int j = 0; j < 8; ++j)
                ep[j] += eak * Wl[k * 256 + lane + 32 * j];
        }
        const float* xr = xp + (long)src * 256;
        float* ar = agg + (long)dst * 256;
        #pragma unroll
        for (int j = 0; j < 8; ++j) {
            int c = lane + 32 * j;
            float v = w4[j >> 1] * ep[j] * xr[c];
            atomicAdd(&ar[c], v);
        }
    }
}

// ---------------------------------------------------------------------------
// GRU gates + LayerNorm, wave per node (2 elems/lane, D=64).
// ---------------------------------------------------------------------------
__global__ __launch_bounds__(256)
void k_gates_ln(const float* __restrict__ gi, const float* __restrict__ gh,
                float* __restrict__ h, const float* __restrict__ ln_g,
                const float* __restrict__ ln_b, float* __restrict__ xout, int Nn)
{
    int node = blockIdx.x * 8 + (threadIdx.x >> 5);
    if (node >= Nn) return;
    int lane = threadIdx.x & 31;
    const float* gir = gi + (long)node * 192;
    const float* ghr = gh + (long)node * 192;
    float* hr = h + (long)node * 64;

    float hn[2];
    #pragma unroll
    for (int j = 0; j < 2; ++j) {
        int d = lane + 32 * j;
        float r = sigmoidf_(gir[d] + ghr[d]);
        float z = sigmoidf_(gir[64 + d] + ghr[64 + d]);
        float n = tanhf(gir[128 + d] + r * ghr[128 + d]);
        hn[j] = (1.0f - z) * n + z * hr[d];
    }
    float sum = hn[0] + hn[1];
    float sq  = hn[0] * hn[0] + hn[1] * hn[1];
    #pragma unroll
    for (int off = 16; off > 0; off >>= 1) {
        sum += __shfl_xor(sum, off, 32);
        sq  += __shfl_xor(sq, off, 32);
    }
    float mu  = sum * (1.0f / 64.0f);
    float var = sq * (1.0f / 64.0f) - mu * mu;
    float rs  = rsqrtf(var + 1e-5f);
    #pragma unroll
    for (int j = 0; j < 2; ++j) {
        int d = lane + 32 * j;
        hr[d] = hn[j];
        xout[(long)node * 64 + d] = (hn[j] - mu) * rs * ln_g[d] + ln_b[d];
    }
}

// ---------------------------------------------------------------------------
extern "C" void kernel_launch(void* const* d_in, const int* in_sizes, int n_in,
                              void* d_out, int out_size, void* d_ws, size_t ws_size,
                              hipStream_t stream)
{
    const float* x       = (const float*)d_in[0];
    const int*   ei      = (const int*)  d_in[1];
    const float* ea      = (const float*)d_in[2];
    const float* W_node  = (const float*)d_in[3];
    const float* W_edge  = (const float*)d_in[4];
    const float* W_att   = (const float*)d_in[5];
    const float* W_scale = (const float*)d_in[6];
    const float* b_scale = (const float*)d_in[7];
    const float* W_ih    = (const float*)d_in[8];
    const float* W_hh    = (const float*)d_in[9];
    const float* b_ih    = (const float*)d_in[10];
    const float* b_hh    = (const float*)d_in[11];
    const float* ln_g    = (const float*)d_in[12];
    const float* ln_b    = (const float*)d_in[13];
    float* out = (float*)d_out;

    const int Nn = in_sizes[0] / D_;        // 50000
    const int En = in_sizes[2] / ED_;       // 800000

    // workspace carve-up (floats)
    float* w = (float*)d_ws;
    float* xp    = w; w += (long)Nn * 256;
    float* agg   = w; w += (long)Nn * 256;
    float* alpha = w; w += (long)En * 4;
    float* amax  = w; w += (long)Nn * 4;
    float* denom = w; w += (long)Nn * 4;
    float* sv    = w; w += (long)Nn * 4;
    float* tv    = w; w += (long)Nn * 4;
    float* gi    = w; w += (long)Nn * 192;
    float* gh    = w; w += (long)Nn * 192;
    float* mbuf  = w; w += (long)Nn * 64;
    float* xcur  = w; w += (long)Nn * 64;
    float* hbuf  = w; w += (long)Nn * 64;
    float* w2e   = w; w += 128;

    long nx = (long)Nn * D_;
    k_init<<<(int)((nx + 255) / 256), 256, 0, stream>>>(x, xcur, hbuf, nx);
    k_w2e<<<1, 128, 0, stream>>>(W_edge, W_att, w2e);

    for (int t = 0; t < 3; ++t) {
        // xp = x @ W_node  [N,64]x[64,256]
        gemm_wmma_f32<false, 0, false><<<dim3(Nn / 16, 256 / 64), 128, 0, stream>>>(
            xcur, W_node, nullptr, xp, Nn, 64, 256);
        k_score_pre<<<(Nn + 7) / 8, 256, 0, stream>>>(xp, W_att, sv, tv, Nn);
        k_reset<<<(int)(((long)Nn * 256 + 255) / 256), 256, 0, stream>>>(agg, amax, denom, Nn);
        k_att<<<(En + 255) / 256, 256, 0, stream>>>(ei, ea, sv, tv, w2e, alpha, amax, En);
        k_exp<<<(En + 255) / 256, 256, 0, stream>>>(ei, alpha, amax, denom, En);
        k_msg<<<(En + MSG_EPB - 1) / MSG_EPB, 256, 0, stream>>>(
            ei, ea, xp, W_edge, alpha, denom, agg, En);
        // m = celu(agg @ W_scale + b_scale)  [N,256]x[256,64]
        gemm_wmma_f32<false, 1, true><<<dim3(Nn / 16, 1), 128, 0, stream>>>(
            agg, W_scale, b_scale, mbuf, Nn, 256, 64);
        // gi = m @ W_ih^T + b_ih ; gh = h @ W_hh^T + b_hh   [N,64]x[64,192]
        gemm_wmma_f32<true, 0, true><<<dim3(Nn / 16, 3), 128, 0, stream>>>(
            mbuf, W_ih, b_ih, gi, Nn, 64, 192);
        gemm_wmma_f32<true, 0, true><<<dim3(Nn / 16, 3), 128, 0, stream>>>(
            hbuf, W_hh, b_hh, gh, Nn, 64, 192);
        float* xo = (t == 2) ? out : xcur;
        k_gates_ln<<<(Nn + 7) / 8, 256, 0, stream>>>(gi, gh, hbuf, ln_g, ln_b, xo, Nn);
    }
}